// RGCN_graph_49727131353505
// MI455X (gfx1250) — compile-verified
//
#include <hip/hip_runtime.h>
#include <hip/hip_bf16.h>

// ---------------- problem constants (from reference) ----------------
#define NN      200000      // nodes
#define NE      3200000     // edges
#define NR      20          // relations
#define FIN     16
#define HID1    16
#define HID2    32
#define NCLS    2
#define NGRAPH  64

typedef float v2f __attribute__((ext_vector_type(2)));
typedef float v8f __attribute__((ext_vector_type(8)));

// ---------------- generic zero ----------------
__global__ __launch_bounds__(256) void k_zero(unsigned int* __restrict__ p, long long n) {
  long long i = (long long)blockIdx.x * blockDim.x + threadIdx.x;
  if (i < n) p[i] = 0u;
}

// ---------------- per-(relation,dst) in-degree ----------------
__global__ __launch_bounds__(256) void k_count(const int* __restrict__ dst,
                                               const int* __restrict__ et,
                                               int* __restrict__ counts) {
  int e = blockIdx.x * blockDim.x + threadIdx.x;
  if (e < NE) atomicAdd(&counts[et[e] * NN + dst[e]], 1);
}

__global__ __launch_bounds__(256) void k_norm(const int* __restrict__ dst,
                                              const int* __restrict__ et,
                                              const int* __restrict__ counts,
                                              float* __restrict__ norm) {
  int e = blockIdx.x * blockDim.x + threadIdx.x;
  if (e < NE) {
    int c = counts[et[e] * NN + dst[e]];
    norm[e] = 1.0f / (float)(c > 1 ? c : 1);
  }
}

// ---------------- root transform via V_WMMA_F32_16X16X4_F32 ----------------
// One wave32 per 16-node tile.  out[node][0 : 16*CT] = X[node][0:16] @ W + bias
// A (16x4 f32) layout per ISA: VGPR0 lanes0-15 -> K=0, lanes16-31 -> K=2;
//                              VGPR1 lanes0-15 -> K=1, lanes16-31 -> K=3.
// B assumed mirrored (K rows half-split like C/D M rows).
// C/D (16x16 f32): VGPR v, lanes0-15 -> (M=v, N=lane); lanes16-31 -> (M=v+8, N=lane-16).
template <int CT>
__global__ __launch_bounds__(256) void k_root_wmma(const float* __restrict__ X,
                                                   const float* __restrict__ W,     // [16][16*CT] row-major
                                                   const float* __restrict__ bias,  // [16*CT]
                                                   float* __restrict__ out) {
  const int wave  = (int)((blockIdx.x * (unsigned)blockDim.x + threadIdx.x) >> 5);
  const int ntile = NN / 16;                 // 12500 exactly
  if (wave >= ntile) return;                 // wave-uniform branch: idle waves never reach WMMA
  const int lane = threadIdx.x & 31;
  const int m    = lane & 15;                // row (A) / column (B,C)
  const int half = lane >> 4;                // 0 or 1
  const int OUT  = 16 * CT;
  const int rowA = wave * 16 + m;

#pragma unroll
  for (int ct = 0; ct < CT; ++ct) {
    v8f c = {};
#pragma unroll
    for (int kk = 0; kk < FIN; kk += 4) {
      const int kb = kk + half * 2;
      v2f a, b;
      a.x = X[rowA * FIN + kb];
      a.y = X[rowA * FIN + kb + 1];
      b.x = W[kb * OUT + ct * 16 + m];
      b.y = W[(kb + 1) * OUT + ct * 16 + m];
      c = __builtin_amdgcn_wmma_f32_16x16x4_f32(false, a, false, b, (short)0, c,
                                                false, false);
    }
    const float bv = bias[ct * 16 + m];
#pragma unroll
    for (int v = 0; v < 8; ++v) {
      const int orow = wave * 16 + v + half * 8;
      out[orow * OUT + ct * 16 + m] = c[v] + bv;
    }
  }
}

// ---------------- per-edge message + scatter (H out channels, K=16 in) ----------------
// One lane per (edge, out-channel).  Relation weights cached in LDS
// (20KB for H=16, 40KB for H=32 -- well under the 320KB WGP LDS).
template <int H>
__global__ __launch_bounds__(256) void k_edge_msg(const float* __restrict__ X,   // [NN][16]
                                                  const float* __restrict__ W,   // [NR][16][H]
                                                  const int* __restrict__ src,
                                                  const int* __restrict__ dst,
                                                  const int* __restrict__ et,
                                                  const float* __restrict__ norm,
                                                  float* __restrict__ acc) {     // [NN][H]
  __shared__ float Wl[NR * FIN * H];
  for (int i = threadIdx.x; i < NR * FIN * H; i += blockDim.x) Wl[i] = W[i];
  __syncthreads();

  const long long gid = (long long)blockIdx.x * blockDim.x + threadIdx.x;
  const int SHIFT = (H == 16) ? 4 : 5;
  const int e = (int)(gid >> SHIFT);
  const int o = (int)(gid & (H - 1));
  if (e >= NE) return;

  const int   r  = et[e];
  const int   s  = src[e];
  const int   d  = dst[e];
  const float nm = norm[e];
  const float* wr = &Wl[r * FIN * H];
  const float4* xv = (const float4*)(X + (long long)s * FIN);  // broadcast across channel group

  float a = 0.f;
#pragma unroll
  for (int kk = 0; kk < 4; ++kk) {
    const float4 xq = xv[kk];
    a += xq.x * wr[(kk * 4 + 0) * H + o];
    a += xq.y * wr[(kk * 4 + 1) * H + o];
    a += xq.z * wr[(kk * 4 + 2) * H + o];
    a += xq.w * wr[(kk * 4 + 3) * H + o];
  }
  atomicAdd(&acc[(long long)d * H + o], a * nm);
}

// ---------------- in-place ReLU ----------------
__global__ __launch_bounds__(256) void k_relu(float* __restrict__ p, long long n) {
  long long i = (long long)blockIdx.x * blockDim.x + threadIdx.x;
  if (i < n) { float v = p[i]; p[i] = v > 0.f ? v : 0.f; }
}

// ---------------- fused ReLU + mean-pool partial sums ----------------
// batch[] is sorted, so each thread run-accumulates RUN consecutive nodes for
// its channel and flushes one atomic per graph change (~30x fewer atomics).
#define RUN 32
__global__ __launch_bounds__(256) void k_pool(const float* __restrict__ h2,
                                              const int* __restrict__ batch,
                                              float* __restrict__ pooled,  // [NGRAPH][HID2]
                                              float* __restrict__ cnt) {   // [NGRAPH]
  const int t = blockIdx.x * blockDim.x + threadIdx.x;
  const int c = t & 31;
  const int n0 = (t >> 5) * RUN;
  if (n0 >= NN) return;
  const int n1 = (n0 + RUN < NN) ? n0 + RUN : NN;

  int curg = batch[n0];
  float s = 0.f, k = 0.f;
  for (int n = n0; n < n1; ++n) {
    const int g = batch[n];
    if (g != curg) {
      atomicAdd(&pooled[curg * HID2 + c], s);
      if (c == 0) atomicAdd(&cnt[curg], k);
      s = 0.f; k = 0.f; curg = g;
    }
    const float v = h2[(long long)n * HID2 + c];
    s += v > 0.f ? v : 0.f;
    k += 1.f;
  }
  atomicAdd(&pooled[curg * HID2 + c], s);
  if (c == 0) atomicAdd(&cnt[curg], k);
}

// ---------------- final FC: out[g][c] = mean_pool(g) @ fcW + fcb ----------------
__global__ __launch_bounds__(128) void k_fc(const float* __restrict__ pooled,
                                            const float* __restrict__ cnt,
                                            const float* __restrict__ fcW,   // [HID2][NCLS]
                                            const float* __restrict__ fcb,   // [NCLS]
                                            float* __restrict__ out) {       // [NGRAPH][NCLS]
  const int t = threadIdx.x;
  if (t >= NGRAPH * NCLS) return;
  const int g = t / NCLS, c = t % NCLS;
  float cn = cnt[g]; cn = cn > 1.f ? cn : 1.f;
  float s = fcb[c];
#pragma unroll
  for (int k = 0; k < HID2; ++k) s += (pooled[g * HID2 + k] / cn) * fcW[k * NCLS + c];
  out[t] = s;
}

// ---------------- launch ----------------
extern "C" void kernel_launch(void* const* d_in, const int* in_sizes, int n_in,
                              void* d_out, int out_size, void* d_ws, size_t ws_size,
                              hipStream_t stream) {
  const float* x     = (const float*)d_in[0];
  const int*   eidx  = (const int*)  d_in[1];   // [2][NE]: row0=src, row1=dst
  const int*   etype = (const int*)  d_in[2];
  const int*   batch = (const int*)  d_in[3];
  const float* W1    = (const float*)d_in[4];   // [NR][16][16]
  const float* root1 = (const float*)d_in[5];   // [16][16]
  const float* b1    = (const float*)d_in[6];
  const float* W2    = (const float*)d_in[7];   // [NR][16][32]
  const float* root2 = (const float*)d_in[8];   // [16][32]
  const float* b2    = (const float*)d_in[9];
  const float* fcW   = (const float*)d_in[10];  // [32][2]
  const float* fcb   = (const float*)d_in[11];
  float* out = (float*)d_out;

  const int* src = eidx;
  const int* dst = eidx + NE;

  // workspace layout (all 4-byte elements; norm placed last: no zeroing needed)
  char* ws = (char*)d_ws;
  long long off = 0;
  int*   counts = (int*)  (ws + off); off += (long long)NR * NN * 4;        // 16.0 MB
  float* h1     = (float*)(ws + off); off += (long long)NN * HID1 * 4;      // 12.8 MB
  float* h2     = (float*)(ws + off); off += (long long)NN * HID2 * 4;      // 25.6 MB
  float* pooled = (float*)(ws + off); off += (long long)NGRAPH * HID2 * 4;
  float* cnt    = (float*)(ws + off); off += (long long)NGRAPH * 4;
  float* norm   = (float*)(ws + off); off += (long long)NE * 4;             // 12.8 MB
  // total ~68 MB, contiguous; counts..cnt must be zeroed each call.

  const long long zn = (long long)NR * NN + (long long)NN * HID1 +
                       (long long)NN * HID2 + NGRAPH * HID2 + NGRAPH;
  k_zero<<<(unsigned)((zn + 255) / 256), 256, 0, stream>>>((unsigned int*)counts, zn);

  // degree counts + per-edge mean norm
  k_count<<<(NE + 255) / 256, 256, 0, stream>>>(dst, etype, counts);
  k_norm <<<(NE + 255) / 256, 256, 0, stream>>>(dst, etype, counts, norm);

  // ----- layer 1 -----
  const int ntile = NN / 16;                         // 12500 wave-tiles
  const int rblocks = (ntile * 32 + 255) / 256;
  k_root_wmma<1><<<rblocks, 256, 0, stream>>>(x, root1, b1, h1);
  k_edge_msg<HID1><<<(int)(((long long)NE * HID1 + 255) / 256), 256, 0, stream>>>(
      x, W1, src, dst, etype, norm, h1);
  k_relu<<<(unsigned)(((long long)NN * HID1 + 255) / 256), 256, 0, stream>>>(
      h1, (long long)NN * HID1);

  // ----- layer 2 -----
  k_root_wmma<2><<<rblocks, 256, 0, stream>>>(h1, root2, b2, h2);
  k_edge_msg<HID2><<<(int)(((long long)NE * HID2 + 255) / 256), 256, 0, stream>>>(
      h1, W2, src, dst, etype, norm, h2);

  // ----- fused ReLU + mean pool + FC -----
  const int pthreads = (NN + RUN - 1) / RUN * 32;
  k_pool<<<(pthreads + 255) / 256, 256, 0, stream>>>(h2, batch, pooled, cnt);
  k_fc<<<1, 128, 0, stream>>>(pooled, cnt, fcW, fcb, out);
}